// Model_9835475108473
// MI455X (gfx1250) — compile-verified
//
#include <hip/hip_runtime.h>
#include <math.h>

// ---------------- problem constants ----------------
#define BB   32
#define TT   512
#define AA   22
#define DD   4
#define HID  64
#define DFF  128
#define PRED 96
#define COUT 44
#define OUTN (BB*PRED*COUT)
#define POOLN (BB*HID)     // pooled sums in ws
#define ZOFF  2048         // float offset of z buffer in ws

typedef float v2f __attribute__((ext_vector_type(2)));
typedef float v8f __attribute__((ext_vector_type(8)));

__device__ __forceinline__ v8f wmma_f32(v2f a, v2f b, v8f c) {
  // V_WMMA_F32_16X16X4_F32 : D(16x16 f32) = A(16x4) * B(4x16) + C
  return __builtin_amdgcn_wmma_f32_16x16x4_f32(
      false, a, false, b, (short)0, c, false, false);
}

// D[32][N] = A[32][4*KSTEPS] @ B[4*KSTEPS][N] (+ bias[n] if HAS_BIAS), N = 16*NTILES.
// A, D in LDS (row-major, lda/ldd floats); B generic (global weights or LDS).
template<int KSTEPS, int NTILES, bool HAS_BIAS>
__device__ __forceinline__ void gemm32(const float* __restrict__ A, int lda,
                                       const float* __restrict__ Bm, int ldb,
                                       float* __restrict__ Dst, int ldd,
                                       const float* __restrict__ bias)
{
  const int lane = threadIdx.x;        // 0..31, wave32
  const int hf   = lane >> 4;          // half-wave select
  const int l16  = lane & 15;
  #pragma unroll
  for (int ni = 0; ni < NTILES; ++ni) {
    v8f acc0 = {};                     // M rows 0..15
    v8f acc1 = {};                     // M rows 16..31
    const int n = ni * 16 + l16;
    #pragma unroll
    for (int ks = 0; ks < KSTEPS; ++ks) {
      const int kb = ks * 4 + hf * 2;  // K base for this half-wave
      v2f bf; bf.x = Bm[(kb + 0) * ldb + n];
              bf.y = Bm[(kb + 1) * ldb + n];
      v2f a0; a0.x = A[l16 * lda + kb];        a0.y = A[l16 * lda + kb + 1];
      v2f a1; a1.x = A[(16 + l16) * lda + kb]; a1.y = A[(16 + l16) * lda + kb + 1];
      acc0 = wmma_f32(a0, bf, acc0);
      acc1 = wmma_f32(a1, bf, acc1);
    }
    if (HAS_BIAS) {
      const float bv = bias[n];
      #pragma unroll
      for (int r = 0; r < 8; ++r) {
        Dst[(r + hf * 8) * ldd + n]      = acc0[r] + bv;   // M = r + hf*8
        Dst[(16 + r + hf * 8) * ldd + n] = acc1[r] + bv;   // M = 16 + r + hf*8
      }
    } else {
      #pragma unroll
      for (int r = 0; r < 8; ++r) {
        Dst[(r + hf * 8) * ldd + n]      = acc0[r];
        Dst[(16 + r + hf * 8) * ldd + n] = acc1[r];
      }
    }
  }
}

// One TGCN layer with H=0: Out = relu((1-sigmoid(Sz)) * tanh(Sh)), rows>=22 zeroed.
template<int KINS>
__device__ __forceinline__ void tgcn_layer(
    const float* __restrict__ In, int ldin,
    const float* czw, const float* czb, const float* lzw, const float* lzb,
    const float* chw, const float* chb, const float* lhw, const float* lhb,
    const float* An, float* T0, float* T1, float* T2, float* Out)
{
  gemm32<KINS, 4, false>(In, ldin, czw, HID, T0, HID, nullptr); __syncthreads(); // XWz
  gemm32<8,    4, true >(An, 32,   T0,  HID, T1, HID, czb);     __syncthreads(); // Gz
  gemm32<16,   4, true >(T1, HID,  lzw, HID, T2, HID, lzb);     __syncthreads(); // Sz
  gemm32<KINS, 4, false>(In, ldin, chw, HID, T0, HID, nullptr); __syncthreads(); // XWh
  gemm32<8,    4, true >(An, 32,   T0,  HID, T1, HID, chb);     __syncthreads(); // Gh
  gemm32<16,   4, true >(T1, HID,  lhw, HID, T0, HID, lhb);     __syncthreads(); // Sh
  const int lane = threadIdx.x;
  for (int idx = lane; idx < 32 * HID; idx += 32) {
    const int i = idx >> 6;
    const float z  = 1.0f / (1.0f + __expf(-T2[idx]));
    const float ht = tanhf(T0[idx]);
    const float v  = (1.0f - z) * ht;
    Out[idx] = (i < AA) ? fmaxf(v, 0.0f) : 0.0f;
  }
  __syncthreads();
}

__global__ __launch_bounds__(32)
void tgcn_main(const float* __restrict__ x,
               const float* czw0, const float* czb0, const float* lzw0, const float* lzb0,
               const float* chw0, const float* chb0, const float* lhw0, const float* lhb0,
               const float* czw1, const float* czb1, const float* lzw1, const float* lzb1,
               const float* chw1, const float* chb1, const float* lhw1, const float* lhb1,
               float* __restrict__ pool)
{
  __shared__ float Xs[32 * DD];
  __shared__ float Ans[32 * 32];
  __shared__ float Dinv[32];
  __shared__ float T0[32 * HID];
  __shared__ float T1[32 * HID];
  __shared__ float T2[32 * HID];
  __shared__ float Hs[32 * HID];

  const int lane = threadIdx.x;
  const int bt   = blockIdx.x;       // b*T + t
  const int b    = bt >> 9;          // T = 512

  // ---- load X tile (22x4 valid, pad to 32x4) ----
  const float* xb = x + (size_t)bt * (AA * DD);
  for (int idx = lane; idx < 32 * DD; idx += 32)
    Xs[idx] = (idx < AA * DD) ? xb[idx] : 0.0f;
  __syncthreads();

  // ---- adjacency: w = 1/(cdist+eps) + I, then D^-1/2 w D^-1/2, padded w/ zeros ----
  for (int idx = lane; idx < 1024; idx += 32) {
    const int i = idx >> 5, j = idx & 31;
    float v = 0.0f;
    if (i < AA && j < AA) {
      float d2 = 0.0f;
      #pragma unroll
      for (int d = 0; d < DD; ++d) {
        const float t = Xs[i * DD + d] - Xs[j * DD + d];
        d2 += t * t;
      }
      v = 1.0f / (sqrtf(d2) + 1e-6f) + ((i == j) ? 1.0f : 0.0f);
    }
    Ans[idx] = v;
  }
  __syncthreads();
  {
    float dv = 0.0f;
    if (lane < AA) {
      float s = 0.0f;
      for (int j = 0; j < AA; ++j) s += Ans[lane * 32 + j];
      dv = rsqrtf(s);
    }
    Dinv[lane] = dv;
  }
  __syncthreads();
  for (int idx = lane; idx < 1024; idx += 32) {
    const int i = idx >> 5, j = idx & 31;
    Ans[idx] *= Dinv[i] * Dinv[j];
  }
  __syncthreads();

  // ---- layer 0 (in: X, K=4) -> Hs ; layer 1 (in: Hs, K=64) -> T1 ----
  tgcn_layer<1 >(Xs, DD,  czw0, czb0, lzw0, lzb0, chw0, chb0, lhw0, lhb0,
                 Ans, T0, T1, T2, Hs);
  tgcn_layer<16>(Hs, HID, czw1, czb1, lzw1, lzb1, chw1, chb1, lhw1, lhb1,
                 Ans, T0, T1, T2, T1);

  // ---- column sums over valid rows -> atomic accumulate into pooled sums ----
  #pragma unroll
  for (int c = 0; c < 2; ++c) {
    const int n = lane + c * 32;
    float s = 0.0f;
    for (int i = 0; i < AA; ++i) s += T1[i * HID + n];
    atomicAdd(&pool[b * HID + n], s);
  }
}

__global__ void zero_kernel(float* p, int n) {
  const int i = blockIdx.x * 256 + threadIdx.x;
  if (i < n) p[i] = 0.0f;
}

__global__ void dec1_kernel(const float* __restrict__ pool,
                            const float* __restrict__ w1,
                            const float* __restrict__ b1,
                            float* __restrict__ zbuf)
{
  const int b = blockIdx.x;        // 0..31
  const int j = threadIdx.x;       // 0..127
  const float invN = 1.0f / (float)(TT * AA);
  float s = b1[j];
  #pragma unroll 8
  for (int h = 0; h < HID; ++h)
    s += (pool[b * HID + h] * invN) * w1[h * DFF + j];
  zbuf[b * DFF + j] = fmaxf(s, 0.0f);
}

__global__ void dec2_kernel(const float* __restrict__ zbuf,
                            const float* __restrict__ w2,
                            const float* __restrict__ b2,
                            float* __restrict__ out)
{
  const int idx = blockIdx.x * 256 + threadIdx.x;
  if (idx >= OUTN) return;
  const int b = idx / (PRED * COUT);
  const int o = idx % (PRED * COUT);
  float s = b2[o];
  #pragma unroll 8
  for (int k = 0; k < DFF; ++k)
    s += zbuf[b * DFF + k] * w2[k * (PRED * COUT) + o];
  out[idx] = s;
}

extern "C" void kernel_launch(void* const* d_in, const int* in_sizes, int n_in,
                              void* d_out, int out_size, void* d_ws, size_t ws_size,
                              hipStream_t stream)
{
  const float* x    = (const float*)d_in[0];
  // layer 0 (r-gate inputs 8..11 unused: H=0 kills the R path)
  const float* czw0 = (const float*)d_in[4];
  const float* czb0 = (const float*)d_in[5];
  const float* lzw0 = (const float*)d_in[6];
  const float* lzb0 = (const float*)d_in[7];
  const float* chw0 = (const float*)d_in[12];
  const float* chb0 = (const float*)d_in[13];
  const float* lhw0 = (const float*)d_in[14];
  const float* lhb0 = (const float*)d_in[15];
  // layer 1 (r-gate inputs 20..23 unused)
  const float* czw1 = (const float*)d_in[16];
  const float* czb1 = (const float*)d_in[17];
  const float* lzw1 = (const float*)d_in[18];
  const float* lzb1 = (const float*)d_in[19];
  const float* chw1 = (const float*)d_in[24];
  const float* chb1 = (const float*)d_in[25];
  const float* lhw1 = (const float*)d_in[26];
  const float* lhb1 = (const float*)d_in[27];
  const float* dw1  = (const float*)d_in[28];
  const float* db1  = (const float*)d_in[29];
  const float* dw2  = (const float*)d_in[30];
  const float* db2  = (const float*)d_in[31];

  float* pool = (float*)d_ws;            // [32*64] pooled sums
  float* zbuf = (float*)d_ws + ZOFF;     // [32*128] decoder hidden
  float* out  = (float*)d_out;

  zero_kernel<<<(POOLN + 255) / 256, 256, 0, stream>>>(pool, POOLN);

  tgcn_main<<<BB * TT, 32, 0, stream>>>(x,
      czw0, czb0, lzw0, lzb0, chw0, chb0, lhw0, lhb0,
      czw1, czb1, lzw1, lzb1, chw1, chb1, lhw1, lhb1, pool);

  dec1_kernel<<<BB, DFF, 0, stream>>>(pool, dw1, db1, zbuf);
  dec2_kernel<<<(OUTN + 255) / 256, 256, 0, stream>>>(zbuf, dw2, db2, out);
}